// MANDONodeClassifier_13881334300974
// MI455X (gfx1250) — compile-verified
//
#include <hip/hip_runtime.h>
#include <hip/hip_bf16.h>
#include <math.h>

// Problem constants (match reference)
constexpr int N   = 50000;
constexpr int E   = 500000;
constexpr int M   = 3;
constexpr int IN  = 8;
constexpr int H   = 8;
constexpr int D   = 32;
constexpr int HD  = H * D;      // 256
constexpr int SEM = 128;
constexpr int OUT = 2;

constexpr int N_TILES  = N / 16;        // 3125
constexpr int C_TILES  = HD / 16;       // 16
constexpr int R_TILES  = (M * N) / 16;  // 9375 semantic row tiles

typedef __attribute__((ext_vector_type(2))) float v2f;
typedef __attribute__((ext_vector_type(8))) float v8f;

// ---------------- utility ----------------

__global__ void k_fill(float* __restrict__ p, int n, float v) {
    int i = blockIdx.x * blockDim.x + threadIdx.x;
    if (i < n) p[i] = v;
}

__global__ void k_fill4(float4* __restrict__ p, int n4, float v) {
    int i = blockIdx.x * blockDim.x + threadIdx.x;
    if (i < n4) p[i] = make_float4(v, v, v, v);
}

__device__ __forceinline__ float atomicMaxFloat(float* addr, float value) {
    if (value >= 0.0f)
        return __int_as_float(atomicMax((int*)addr, __float_as_int(value)));
    else
        return __uint_as_float(atomicMin((unsigned int*)addr, __float_as_uint(value)));
}

__device__ __forceinline__ float leaky02(float x) { return x > 0.0f ? x : 0.2f * x; }
__device__ __forceinline__ float eluf(float x)    { return x > 0.0f ? x : (__expf(x) - 1.0f); }

// ---------------- 1) feat = h @ fc_w  (f32 WMMA, K=8 -> two 16x16x4 steps) ----------------
// grid: 37500 blocks x 128 threads (4 waves/block), 150000 tiles total
__global__ __launch_bounds__(128) void k_feat_wmma(const float* __restrict__ h,
                                                   const float* __restrict__ fcw,
                                                   float* __restrict__ feat) {
    const int wave = threadIdx.x >> 5;
    const int lane = threadIdx.x & 31;
    const int tile = blockIdx.x * 4 + wave;          // 0 .. 149999
    const int nt   = tile % N_TILES;
    const int rest = tile / N_TILES;
    const int ct   = rest % C_TILES;
    const int m    = rest / C_TILES;

    const int n0 = nt * 16;
    const int c0 = ct * 16;
    const float* W = fcw + (size_t)m * IN * HD;

    const int hr   = lane & 15;              // A row / B col within tile
    const int koff = (lane < 16) ? 0 : 2;    // K sub-offset per half-wave

    v8f acc = {};
    #pragma unroll
    for (int k0 = 0; k0 < IN; k0 += 4) {
        v2f a, b;
        const int kb = k0 + koff;
        a.x = h[(size_t)(n0 + hr) * IN + kb];
        a.y = h[(size_t)(n0 + hr) * IN + kb + 1];
        b.x = W[(size_t)kb * HD + c0 + hr];
        b.y = W[(size_t)(kb + 1) * HD + c0 + hr];
        acc = __builtin_amdgcn_wmma_f32_16x16x4_f32(false, a, false, b,
                                                    (short)0, acc, false, false);
    }

    float* out = feat + ((size_t)m * N + n0) * HD + c0;
    const int rbase = (lane < 16) ? 0 : 8;
    #pragma unroll
    for (int r = 0; r < 8; ++r)
        out[(size_t)(rbase + r) * HD + hr] = acc[r];
}

// ---------------- 2) el/er per (m, n, head) — float4 reads ----------------
__global__ void k_elr(const float* __restrict__ feat,
                      const float* __restrict__ attn_l,
                      const float* __restrict__ attn_r,
                      float* __restrict__ el, float* __restrict__ er) {
    int idx = blockIdx.x * blockDim.x + threadIdx.x;
    if (idx >= M * N * H) return;
    const int hh = idx % H;
    const int n  = (idx / H) % N;
    const int m  = idx / (H * N);
    const float4* f  = (const float4*)(feat + ((size_t)m * N + n) * HD + hh * D);
    const float4* al = (const float4*)(attn_l + ((size_t)m * H + hh) * D);
    const float4* ar = (const float4*)(attn_r + ((size_t)m * H + hh) * D);
    float sl = 0.0f, sr = 0.0f;
    #pragma unroll
    for (int q = 0; q < D / 4; ++q) {
        const float4 fv = f[q], lv = al[q], rv = ar[q];
        sl += fv.x * lv.x + fv.y * lv.y + fv.z * lv.z + fv.w * lv.w;
        sr += fv.x * rv.x + fv.y * rv.y + fv.z * rv.z + fv.w * rv.w;
    }
    el[idx] = sl;
    er[idx] = sr;
}

// ---------------- helper: gather 8 heads worth of el/er as 2x float4 ----------------
__device__ __forceinline__ void load8(const float* __restrict__ p, float v[8]) {
    const float4 a = ((const float4*)p)[0];
    const float4 b = ((const float4*)p)[1];
    v[0] = a.x; v[1] = a.y; v[2] = a.z; v[3] = a.w;
    v[4] = b.x; v[5] = b.y; v[6] = b.z; v[7] = b.w;
}

// ---------------- 3) edge softmax: max pass (one thread per edge) ----------------
__global__ void k_edge_max(const int* __restrict__ src, const int* __restrict__ dst,
                           const float* __restrict__ el, const float* __restrict__ er,
                           float* __restrict__ nmax) {
    int idx = blockIdx.x * blockDim.x + threadIdx.x;   // m*E + e
    if (idx >= M * E) return;
    const int m = idx / E;
    const int s = src[idx];
    const int d = dst[idx];
    float l[8], r[8];
    load8(el + ((size_t)m * N + s) * H, l);
    load8(er + ((size_t)m * N + d) * H, r);
    float* nm = nmax + ((size_t)m * N + d) * H;
    #pragma unroll
    for (int hh = 0; hh < H; ++hh)
        atomicMaxFloat(&nm[hh], leaky02(l[hh] + r[hh]));
}

// ---------------- 4) edge softmax: denom pass (one thread per edge) ----------------
__global__ void k_edge_sum(const int* __restrict__ src, const int* __restrict__ dst,
                           const float* __restrict__ el, const float* __restrict__ er,
                           const float* __restrict__ nmax, float* __restrict__ denom) {
    int idx = blockIdx.x * blockDim.x + threadIdx.x;
    if (idx >= M * E) return;
    const int m = idx / E;
    const int s = src[idx];
    const int d = dst[idx];
    float l[8], r[8], mx[8];
    load8(el + ((size_t)m * N + s) * H, l);
    load8(er + ((size_t)m * N + d) * H, r);
    load8(nmax + ((size_t)m * N + d) * H, mx);
    float* dn = denom + ((size_t)m * N + d) * H;
    #pragma unroll
    for (int hh = 0; hh < H; ++hh)
        atomicAdd(&dn[hh], __expf(leaky02(l[hh] + r[hh]) - mx[hh]));
}

// ---------------- 5) alpha-weighted scatter-add (one thread per edge) ----------------
__global__ void k_edge_aggr(const int* __restrict__ src, const int* __restrict__ dst,
                            const float* __restrict__ el, const float* __restrict__ er,
                            const float* __restrict__ nmax, const float* __restrict__ denom,
                            const float* __restrict__ feat, float* __restrict__ zacc) {
    int idx = blockIdx.x * blockDim.x + threadIdx.x;
    if (idx >= M * E) return;
    const int m = idx / E;
    const int s = src[idx];
    const int d = dst[idx];

    const float4* fs = (const float4*)(feat + ((size_t)m * N + s) * HD);
    __builtin_prefetch(fs, 0, 1);     // global_prefetch_b8: pull gather line toward L2

    float l[8], r[8], mx[8], dn[8], alpha[8];
    load8(el + ((size_t)m * N + s) * H, l);
    load8(er + ((size_t)m * N + d) * H, r);
    load8(nmax + ((size_t)m * N + d) * H, mx);
    load8(denom + ((size_t)m * N + d) * H, dn);
    #pragma unroll
    for (int hh = 0; hh < H; ++hh)
        alpha[hh] = __expf(leaky02(l[hh] + r[hh]) - mx[hh]) / dn[hh];

    float* zd = zacc + ((size_t)m * N + d) * HD;
    #pragma unroll 8
    for (int q = 0; q < HD / 4; ++q) {          // 64 float4 loads, 256 atomics
        const float4 f = fs[q];
        const float a = alpha[q >> 3];          // 8 float4s per head
        const int c = q * 4;
        atomicAdd(&zd[c + 0], f.x * a);
        atomicAdd(&zd[c + 1], f.y * a);
        atomicAdd(&zd[c + 2], f.z * a);
        atomicAdd(&zd[c + 3], f.w * a);
    }
}

// ---------------- 6) bias + ELU (in place on z) ----------------
__global__ void k_bias_elu(float* __restrict__ z, const float* __restrict__ gat_bias) {
    int idx = blockIdx.x * blockDim.x + threadIdx.x;
    if (idx >= M * N * HD) return;
    const int c = idx % HD;
    const int m = idx / (N * HD);
    z[idx] = eluf(z[idx] + gat_bias[m * HD + c]);
}

// ---------------- 7) semantic attention logits (f32 WMMA, K=256) ----------------
// One wave per 16-row tile of flattened [M*N] rows. z-tile staged in LDS via float4,
// 8 column tiles x 64 K-steps of v_wmma_f32_16x16x4_f32, then
// tanh(+b1)*w2 dot with cross-lane reduction, atomicAdd into wsum[m].
__global__ __launch_bounds__(32) void k_sem_wmma(const float* __restrict__ z,
                                                 const float* __restrict__ w1,
                                                 const float* __restrict__ b1,
                                                 const float* __restrict__ w2,
                                                 float* __restrict__ wsum) {
    __shared__ __align__(16) float lds[16 * HD];   // 16 rows x 256 cols = 16 KB
    const int lane = threadIdx.x;    // 32 threads = 1 wave
    const int r0   = blockIdx.x * 16;
    const int m    = r0 / N;         // tiles never straddle a metapath (N % 16 == 0)

    // stage 16 z rows into LDS (coalesced b128)
    const float4* zrow = (const float4*)(z + (size_t)r0 * HD);
    float4* l4 = (float4*)lds;
    #pragma unroll
    for (int i = lane; i < 16 * HD / 4; i += 32) l4[i] = zrow[i];
    __syncthreads();

    const int hr   = lane & 15;
    const int koff = (lane < 16) ? 0 : 2;

    float pr[8];
    #pragma unroll
    for (int r = 0; r < 8; ++r) pr[r] = 0.0f;

    for (int ctile = 0; ctile < SEM / 16; ++ctile) {   // 8 column tiles
        const int c0 = ctile * 16;
        v8f acc = {};
        for (int k0 = 0; k0 < HD; k0 += 4) {           // 64 K-steps
            v2f a, b;
            const int kb = k0 + koff;
            a.x = lds[hr * HD + kb];
            a.y = lds[hr * HD + kb + 1];
            b.x = w1[(size_t)kb * SEM + c0 + hr];
            b.y = w1[(size_t)(kb + 1) * SEM + c0 + hr];
            acc = __builtin_amdgcn_wmma_f32_16x16x4_f32(false, a, false, b,
                                                        (short)0, acc, false, false);
        }
        const int   col = c0 + hr;
        const float bb  = b1[col];
        const float ww  = w2[col];
        #pragma unroll
        for (int r = 0; r < 8; ++r)
            pr[r] += tanhf(acc[r] + bb) * ww;
    }

    // reduce each row's partial across the 16 lanes of each half-wave
    #pragma unroll
    for (int r = 0; r < 8; ++r) {
        float v = pr[r];
        v += __shfl_xor(v, 8, 16);
        v += __shfl_xor(v, 4, 16);
        v += __shfl_xor(v, 2, 16);
        v += __shfl_xor(v, 1, 16);
        pr[r] = v;
    }
    if ((lane & 15) == 0) {   // lane 0 covers rows 0..7, lane 16 covers rows 8..15
        float s = 0.0f;
        #pragma unroll
        for (int r = 0; r < 8; ++r) s += pr[r];
        atomicAdd(&wsum[m], s);
    }
}

// ---------------- 8) beta = softmax(wsum / N) ----------------
__global__ void k_beta(const float* __restrict__ wsum, float* __restrict__ beta) {
    float b[M];
    float mx = -INFINITY;
    for (int m = 0; m < M; ++m) { b[m] = wsum[m] / (float)N; mx = fmaxf(mx, b[m]); }
    float s = 0.0f;
    for (int m = 0; m < M; ++m) { b[m] = __expf(b[m] - mx); s += b[m]; }
    for (int m = 0; m < M; ++m) beta[m] = b[m] / s;
}

// ---------------- 9) fused = sum_m beta*z ; out = fused @ cls_w + cls_b ----------------
__global__ void k_fuse(const float* __restrict__ z, const float* __restrict__ beta,
                       const float* __restrict__ cw, const float* __restrict__ cb,
                       float* __restrict__ out) {
    int n = blockIdx.x * blockDim.x + threadIdx.x;
    if (n >= N) return;
    const float b0 = beta[0], b1_ = beta[1], b2 = beta[2];
    const float4* z0 = (const float4*)(z + ((size_t)0 * N + n) * HD);
    const float4* z1 = (const float4*)(z + ((size_t)1 * N + n) * HD);
    const float4* z2 = (const float4*)(z + ((size_t)2 * N + n) * HD);
    float a0 = cb[0], a1 = cb[1];
    for (int q = 0; q < HD / 4; ++q) {
        const float4 v0 = z0[q], v1 = z1[q], v2 = z2[q];
        const float f0 = b0 * v0.x + b1_ * v1.x + b2 * v2.x;
        const float f1 = b0 * v0.y + b1_ * v1.y + b2 * v2.y;
        const float f2 = b0 * v0.z + b1_ * v1.z + b2 * v2.z;
        const float f3 = b0 * v0.w + b1_ * v1.w + b2 * v2.w;
        const int c = q * 4;
        a0 += f0 * cw[(c + 0) * OUT] + f1 * cw[(c + 1) * OUT]
            + f2 * cw[(c + 2) * OUT] + f3 * cw[(c + 3) * OUT];
        a1 += f0 * cw[(c + 0) * OUT + 1] + f1 * cw[(c + 1) * OUT + 1]
            + f2 * cw[(c + 2) * OUT + 1] + f3 * cw[(c + 3) * OUT + 1];
    }
    out[(size_t)n * OUT + 0] = a0;
    out[(size_t)n * OUT + 1] = a1;
}

// ---------------- launch ----------------
extern "C" void kernel_launch(void* const* d_in, const int* in_sizes, int n_in,
                              void* d_out, int out_size, void* d_ws, size_t ws_size,
                              hipStream_t stream) {
    const float* h       = (const float*)d_in[0];
    const int*   src     = (const int*)  d_in[1];
    const int*   dst     = (const int*)  d_in[2];
    const float* fc_w    = (const float*)d_in[3];
    const float* attn_l  = (const float*)d_in[4];
    const float* attn_r  = (const float*)d_in[5];
    const float* gatb    = (const float*)d_in[6];
    const float* sem_w1  = (const float*)d_in[7];
    const float* sem_b1  = (const float*)d_in[8];
    const float* sem_w2  = (const float*)d_in[9];
    const float* cls_w   = (const float*)d_in[10];
    const float* cls_b   = (const float*)d_in[11];
    float* out = (float*)d_out;

    // workspace layout (floats)
    float* ws    = (float*)d_ws;
    float* feat  = ws;                               // M*N*HD
    float* z     = feat  + (size_t)M * N * HD;       // M*N*HD (rst accumulator, then z)
    float* el    = z     + (size_t)M * N * HD;       // M*N*H
    float* er    = el    + (size_t)M * N * H;
    float* nmax  = er    + (size_t)M * N * H;
    float* denom = nmax  + (size_t)M * N * H;
    float* wsum  = denom + (size_t)M * N * H;        // M
    float* beta  = wsum  + M;                        // M

    const int TB = 256;
    const int nMNHD = M * N * HD;   // 38,400,000
    const int nMNH  = M * N * H;    //  1,200,000
    const int nME   = M * E;        //  1,500,000

    // init
    k_fill4<<<(nMNHD / 4 + TB - 1) / TB, TB, 0, stream>>>((float4*)z, nMNHD / 4, 0.0f);
    k_fill<<<(nMNH + TB - 1) / TB, TB, 0, stream>>>(nmax, nMNH, -INFINITY);
    k_fill<<<(nMNH + TB - 1) / TB, TB, 0, stream>>>(denom, nMNH, 0.0f);
    k_fill<<<1, 32, 0, stream>>>(wsum, M, 0.0f);

    // GAT per metapath (all metapaths fused via grid dims)
    k_feat_wmma<<<(N_TILES * C_TILES * M) / 4, 128, 0, stream>>>(h, fc_w, feat);
    k_elr<<<(nMNH + TB - 1) / TB, TB, 0, stream>>>(feat, attn_l, attn_r, el, er);
    k_edge_max<<<(nME + TB - 1) / TB, TB, 0, stream>>>(src, dst, el, er, nmax);
    k_edge_sum<<<(nME + TB - 1) / TB, TB, 0, stream>>>(src, dst, el, er, nmax, denom);
    k_edge_aggr<<<(nME + TB - 1) / TB, TB, 0, stream>>>(src, dst, el, er, nmax, denom, feat, z);
    k_bias_elu<<<(nMNHD + TB - 1) / TB, TB, 0, stream>>>(z, gatb);

    // semantic attention + classifier
    k_sem_wmma<<<R_TILES, 32, 0, stream>>>(z, sem_w1, sem_b1, sem_w2, wsum);
    k_beta<<<1, 1, 0, stream>>>(wsum, beta);
    k_fuse<<<(N + TB - 1) / TB, TB, 0, stream>>>(z, beta, cls_w, cls_b, out);
}